// Memory_36541581754966
// MI455X (gfx1250) — compile-verified
//
#include <hip/hip_runtime.h>
#include <hip/hip_bf16.h>

#define BB 32
#define NN 1024
#define WW 128
#define RR 4
#define EPSF 1e-8f

typedef __attribute__((ext_vector_type(2))) float v2f;
typedef __attribute__((ext_vector_type(8))) float v8f;

// ---------------- reduction helpers (wave32) ----------------
__device__ inline float wave_sum(float v) {
    #pragma unroll
    for (int off = 16; off > 0; off >>= 1) v += __shfl_down(v, off, 32);
    return v;
}
__device__ inline float wave_max(float v) {
    #pragma unroll
    for (int off = 16; off > 0; off >>= 1) v = fmaxf(v, __shfl_down(v, off, 32));
    return v;
}
__device__ inline float block_sum(float v, float* red) { // blockDim.x == 1024
    int lane = threadIdx.x & 31, wv = threadIdx.x >> 5;
    v = wave_sum(v);
    if (lane == 0) red[wv] = v;
    __syncthreads();
    if (wv == 0) {
        float r = red[lane];
        r = wave_sum(r);
        if (lane == 0) red[0] = r;
    }
    __syncthreads();
    float out = red[0];
    __syncthreads();
    return out;
}
__device__ inline float block_max(float v, float* red) {
    int lane = threadIdx.x & 31, wv = threadIdx.x >> 5;
    v = wave_max(v);
    if (lane == 0) red[wv] = v;
    __syncthreads();
    if (wv == 0) {
        float r = red[lane];
        r = wave_max(r);
        if (lane == 0) red[0] = r;
    }
    __syncthreads();
    float out = red[0];
    __syncthreads();
    return out;
}

// ---------------- kernel 0: zero accumulators ----------------
__global__ void k_zero(float* p, int n) {
    int i = blockIdx.x * blockDim.x + threadIdx.x;
    if (i < n) p[i] = 0.f;
}

// ---------------- kernel 1: write-head (ww, sum_ww) ----------------
__global__ __launch_bounds__(1024) void k_write_head(
        const float* __restrict__ wk, const float* __restrict__ wstr,
        const float* __restrict__ ag, const float* __restrict__ wg,
        const float* __restrict__ mem, const float* __restrict__ usage,
        float* __restrict__ ww_out, float* __restrict__ sum_ww) {
    int b = blockIdx.x, t = threadIdx.x, lane = t & 31, wv = t >> 5;
    __shared__ float s_cos[NN];
    __shared__ float s_val[NN];
    __shared__ int   s_idx[NN];
    __shared__ float s_alloc[NN];
    __shared__ __align__(16) float s_wk[WW];
    __shared__ float s_red[32];
    __shared__ float s_knorm;

    if (t < WW) s_wk[t] = wk[b * WW + t];
    __syncthreads();
    if (t < 32) {
        float sq = 0.f;
        for (int w = lane; w < WW; w += 32) { float x = s_wk[w]; sq += x * x; }
        sq = wave_sum(sq);
        if (lane == 0) s_knorm = fmaxf(sqrtf(sq), EPSF);
    }
    __syncthreads();

    // cosine vs each old-memory row: one wave per row
    for (int it = 0; it < NN / 32; ++it) {
        int n = it * 32 + wv;
        const float4* mrow = (const float4*)(mem + ((size_t)b * NN + n) * WW);
        float4 m = mrow[lane];
        float4 k = ((const float4*)s_wk)[lane];
        float dot = m.x * k.x + m.y * k.y + m.z * k.z + m.w * k.w;
        float sq  = m.x * m.x + m.y * m.y + m.z * m.z + m.w * m.w;
        dot = wave_sum(dot);
        sq  = wave_sum(sq);
        if (lane == 0) s_cos[n] = dot / (s_knorm * fmaxf(sqrtf(sq), EPSF));
    }
    __syncthreads();

    // softmax over N
    float c  = s_cos[t];
    float mx = block_max(c, s_red);
    float ex = expf(c - mx);
    float sm = block_sum(ex, s_red);
    float wc = (ex / sm) * wstr[b];

    // bitonic sort of usage (value, index) ascending, stable via index tiebreak
    s_val[t] = usage[b * NN + t];
    s_idx[t] = t;
    __syncthreads();
    for (int k = 2; k <= NN; k <<= 1) {
        for (int j = k >> 1; j > 0; j >>= 1) {
            int ixj = t ^ j;
            if (ixj > t) {
                bool up = ((t & k) == 0);
                float a = s_val[t], bb2 = s_val[ixj];
                int ia = s_idx[t], ib = s_idx[ixj];
                bool gt = (a > bb2) || (a == bb2 && ia > ib);
                if (up ? gt : !gt) {
                    s_val[t] = bb2; s_val[ixj] = a;
                    s_idx[t] = ib;  s_idx[ixj] = ia;
                }
            }
            __syncthreads();
        }
    }
    // inclusive cumprod (Hillis-Steele) then exclusive via divide
    float sv = s_val[t];
    s_alloc[t] = sv;
    __syncthreads();
    for (int off = 1; off < NN; off <<= 1) {
        float mine  = s_alloc[t];
        float other = (t >= off) ? s_alloc[t - off] : 1.0f;
        __syncthreads();
        s_alloc[t] = mine * other;
        __syncthreads();
    }
    float cum = s_alloc[t];
    float alloc_sorted = (1.0f - sv) * (cum / sv);
    __syncthreads();
    s_alloc[s_idx[t]] = alloc_sorted;   // scatter back to original order
    __syncthreads();

    float al  = s_alloc[t];
    float www = (ag[b] * (al - wc) + wc) * wg[b];
    ww_out[b * NN + t] = www;
    float tot = block_sum(www, s_red);
    if (t == 0) atomicAdd(sum_ww, tot);
}

// ---------------- kernel 2: memory erase/write ----------------
__global__ __launch_bounds__(256) void k_mem_update(
        const float* __restrict__ mem, const float* __restrict__ ev,
        const float* __restrict__ wvec, const float* __restrict__ ww,
        float* __restrict__ mem_out) {
    int wv = threadIdx.x >> 5, lane = threadIdx.x & 31;
    int g = blockIdx.x * 8 + wv;          // global row id
    int b = g >> 10, n = g & (NN - 1);
    float wwn = ww[b * NN + n];
    const float4* mrow = (const float4*)(mem  + ((size_t)b * NN + n) * WW);
    const float4* e4   = (const float4*)(ev   + (size_t)b * WW);
    const float4* v4   = (const float4*)(wvec + (size_t)b * WW);
    float4 m = mrow[lane], e = e4[lane], v = v4[lane], o;
    o.x = m.x * (1.f - wwn * e.x) + wwn * v.x;
    o.y = m.y * (1.f - wwn * e.y) + wwn * v.y;
    o.z = m.z * (1.f - wwn * e.z) + wwn * v.z;
    o.w = m.w * (1.f - wwn * e.w) + wwn * v.w;
    ((float4*)(mem_out + ((size_t)b * NN + n) * WW))[lane] = o;
}

// ---------------- kernel 3: temporal link + fw + bw (WMMA) ----------------
#define TCOLS 128
__global__ __launch_bounds__(256) void k_link(
        const float* __restrict__ L_in, const float* __restrict__ ww,
        const float* __restrict__ prec, const float* __restrict__ rw,
        float* __restrict__ L_out, float* __restrict__ fw, float* __restrict__ bw) {
    int b = blockIdx.x >> 6;              // N/16 = 64 row-chunks per batch
    int row0 = (blockIdx.x & 63) * 16;
    int t = threadIdx.x, lane = t & 31, wv = t >> 5;
    __shared__ float s_ww[NN];
    __shared__ float s_p[NN];
    __shared__ float s_rw[RR * NN];
    __shared__ float s_lp[16][TCOLS + 4];

    for (int i = t; i < NN; i += 256) { s_ww[i] = ww[b * NN + i]; s_p[i] = prec[b * NN + i]; }
    for (int i = t; i < RR * NN; i += 256) s_rw[i] = rw[(size_t)b * RR * NN + i];
    __syncthreads();

    int r = lane & 15, khalf = lane >> 4;
    bool rvalid = (r < RR);

    // A operand (16x16 rw slice over this block's 16 rows) is loop-invariant:
    // build the 4 K-chunk operands once. Unconditional LDS loads + VALU select
    // (avoids divergent ds_load under exec-mask toggling).
    v2f aop[4];
    #pragma unroll
    for (int kk = 0; kk < 4; ++kk) {
        int k0 = 4 * kk + 2 * khalf;
        float ax = s_rw[(r & 3) * NN + row0 + k0];
        float ay = s_rw[(r & 3) * NN + row0 + k0 + 1];
        aop[kk].x = rvalid ? ax : 0.f;
        aop[kk].y = rvalid ? ay : 0.f;
    }

    float facc0[RR] = {0.f, 0.f, 0.f, 0.f};   // fw for row 2*wv
    float facc1[RR] = {0.f, 0.f, 0.f, 0.f};   // fw for row 2*wv+1

    for (int c0 = 0; c0 < NN; c0 += TCOLS) {
        // build L' tile (16 x TCOLS), write-through to global
        for (int e = t; e < 16 * TCOLS; e += 256) {
            int i = e >> 7, j = e & (TCOLS - 1);
            float wr = s_ww[row0 + i], wcl = s_ww[c0 + j];
            size_t gi = ((size_t)(b * NN + row0 + i)) * NN + c0 + j;
            float lp = (1.f - wr - wcl) * L_in[gi] + wr * s_p[c0 + j];
            s_lp[i][j] = lp;
            L_out[gi] = lp;
        }
        __syncthreads();

        // fw partials: wave wv owns rows 2*wv, 2*wv+1 (full dot along columns = n)
        {
            int i0 = 2 * wv, i1 = 2 * wv + 1;
            for (int j = lane; j < TCOLS; j += 32) {
                int n = c0 + j;
                float l0 = s_lp[i0][j], l1 = s_lp[i1][j];
                #pragma unroll
                for (int rr2 = 0; rr2 < RR; ++rr2) {
                    float rv = s_rw[rr2 * NN + n];
                    facc0[rr2] += rv * l0;
                    facc1[rr2] += rv * l1;
                }
            }
        }

        // bw partials via WMMA: wave wv handles column tile [c0+16*wv, +16)
        v8f cacc;
        #pragma unroll
        for (int i = 0; i < 8; ++i) cacc[i] = 0.f;
        #pragma unroll
        for (int kk = 0; kk < 4; ++kk) {
            int k0 = 4 * kk + 2 * khalf;
            v2f bm;
            bm.x = s_lp[k0][16 * wv + r];
            bm.y = s_lp[k0 + 1][16 * wv + r];
            cacc = __builtin_amdgcn_wmma_f32_16x16x4_f32(
                false, aop[kk], false, bm, (short)0, cacc, false, false);
        }
        if (lane < 16) {
            #pragma unroll
            for (int i = 0; i < RR; ++i)
                atomicAdd(&bw[((size_t)b * RR + i) * NN + c0 + 16 * wv + lane], cacc[i]);
        }
        __syncthreads();
    }

    // finalize fw for this block's rows
    #pragma unroll
    for (int rr2 = 0; rr2 < RR; ++rr2) {
        float v0 = wave_sum(facc0[rr2]);
        float v1 = wave_sum(facc1[rr2]);
        if (lane == 0) {
            fw[((size_t)b * RR + rr2) * NN + row0 + 2 * wv]     = v0;
            fw[((size_t)b * RR + rr2) * NN + row0 + 2 * wv + 1] = v1;
        }
    }
}

// ---------------- kernel 4: read content + combine + usage + precedence ----------------
__global__ __launch_bounds__(1024) void k_read_combine(
        const float* __restrict__ rk, const float* __restrict__ rstr,
        const float* __restrict__ modes, const float* __restrict__ fg,
        const float* __restrict__ mem_new, const float* __restrict__ usage,
        const float* __restrict__ ww, const float* __restrict__ prec,
        const float* __restrict__ fw, const float* __restrict__ bw,
        const float* __restrict__ sum_ww,
        float* __restrict__ rw_out, float* __restrict__ usage_out,
        float* __restrict__ prec_out) {
    int b = blockIdx.x, t = threadIdx.x, lane = t & 31, wv = t >> 5;
    __shared__ float s_cos[RR][NN];           // 16KB
    __shared__ __align__(16) float s_rk[RR][WW];
    __shared__ float s_knorm[RR];
    __shared__ float s_red[32];

    if (t < RR * WW) s_rk[t >> 7][t & (WW - 1)] = rk[(size_t)b * RR * WW + t];
    __syncthreads();
    if (t < 32) {
        for (int rr2 = 0; rr2 < RR; ++rr2) {
            float sq = 0.f;
            for (int w = lane; w < WW; w += 32) { float x = s_rk[rr2][w]; sq += x * x; }
            sq = wave_sum(sq);
            if (lane == 0) s_knorm[rr2] = fmaxf(sqrtf(sq), EPSF);
        }
    }
    __syncthreads();

    // cosine vs NEW memory; norms fused in the same pass (wave per row)
    for (int it = 0; it < NN / 32; ++it) {
        int n = it * 32 + wv;
        const float4* mrow = (const float4*)(mem_new + ((size_t)b * NN + n) * WW);
        float4 m = mrow[lane];
        float sq = m.x * m.x + m.y * m.y + m.z * m.z + m.w * m.w;
        float d[RR];
        #pragma unroll
        for (int rr2 = 0; rr2 < RR; ++rr2) {
            float4 k = ((const float4*)s_rk[rr2])[lane];
            d[rr2] = m.x * k.x + m.y * k.y + m.z * k.z + m.w * k.w;
        }
        sq = wave_sum(sq);
        #pragma unroll
        for (int rr2 = 0; rr2 < RR; ++rr2) d[rr2] = wave_sum(d[rr2]);
        if (lane == 0) {
            float inv = 1.f / fmaxf(sqrtf(sq), EPSF);
            #pragma unroll
            for (int rr2 = 0; rr2 < RR; ++rr2)
                s_cos[rr2][n] = d[rr2] * inv / s_knorm[rr2];
        }
    }
    __syncthreads();

    // per-head softmax over N
    for (int rr2 = 0; rr2 < RR; ++rr2) {
        float c  = s_cos[rr2][t];
        float mx = block_max(c, s_red);
        float ex = expf(c - mx);
        float sm = block_sum(ex, s_red);
        s_cos[rr2][t] = ex / sm;
        __syncthreads();
    }

    float S = *sum_ww;
    float wwn = ww[b * NN + t];
    float prod = 1.f;
    for (int rr2 = 0; rr2 < RR; ++rr2) {
        float cw = s_cos[rr2][t] * rstr[b * RR + rr2];
        const float* md = modes + ((size_t)b * RR + rr2) * 3;
        size_t o = ((size_t)b * RR + rr2) * NN + t;
        float rwv = fw[o] * md[0] + bw[o] * md[1] + cw * md[2];
        rw_out[o] = rwv;
        prod *= rwv;
    }
    float u = usage[b * NN + t];
    float retention = 1.f - fg[b * NN + t] * prod;
    usage_out[b * NN + t] = (u + wwn - u * wwn) * retention;
    prec_out[b * NN + t]  = (1.f - S) * prec[b * NN + t] + wwn;
}

// ---------------- kernel 5: read_val = rw' x mem' (WMMA) ----------------
__global__ __launch_bounds__(256) void k_read_val(
        const float* __restrict__ rw_new, const float* __restrict__ mem_new,
        float* __restrict__ out) {
    int b = blockIdx.x, t = threadIdx.x, lane = t & 31, wv = t >> 5;
    __shared__ float s_rw[RR * NN];
    for (int i = t; i < RR * NN; i += 256) s_rw[i] = rw_new[(size_t)b * RR * NN + i];
    __syncthreads();

    int r = lane & 15, khalf = lane >> 4;
    bool rvalid = (r < RR);
    int w0 = 16 * wv;                      // 8 waves cover W=128
    const float* mb = mem_new + (size_t)b * NN * WW;
    v8f c;
    #pragma unroll
    for (int i = 0; i < 8; ++i) c[i] = 0.f;

    for (int n0 = 0; n0 < NN; n0 += 4) {
        int k0 = n0 + 2 * khalf;
        v2f a, bm;
        // unconditional LDS loads + VALU select (no divergent ds_load)
        float ax = s_rw[(r & 3) * NN + k0];
        float ay = s_rw[(r & 3) * NN + k0 + 1];
        a.x = rvalid ? ax : 0.f;
        a.y = rvalid ? ay : 0.f;
        bm.x = mb[(size_t)k0 * WW + w0 + r];
        bm.y = mb[(size_t)(k0 + 1) * WW + w0 + r];
        c = __builtin_amdgcn_wmma_f32_16x16x4_f32(
            false, a, false, bm, (short)0, c, false, false);
    }
    if (lane < 16) {
        #pragma unroll
        for (int i = 0; i < RR; ++i)
            out[((size_t)b * RR + i) * WW + w0 + lane] = c[i];
    }
}

// ---------------- host launcher ----------------
extern "C" void kernel_launch(void* const* d_in, const int* in_sizes, int n_in,
                              void* d_out, int out_size, void* d_ws, size_t ws_size,
                              hipStream_t stream) {
    const float* wk    = (const float*)d_in[0];
    const float* wstr  = (const float*)d_in[1];
    const float* ag    = (const float*)d_in[2];
    const float* wg    = (const float*)d_in[3];
    const float* wvec  = (const float*)d_in[4];
    const float* ev    = (const float*)d_in[5];
    const float* rk    = (const float*)d_in[6];
    const float* rstr  = (const float*)d_in[7];
    const float* modes = (const float*)d_in[8];
    const float* fg    = (const float*)d_in[9];
    const float* mem   = (const float*)d_in[10];
    const float* usage = (const float*)d_in[11];
    const float* rw    = (const float*)d_in[12];
    const float* L_in  = (const float*)d_in[13];
    const float* prec  = (const float*)d_in[14];

    float* out = (float*)d_out;
    // output regions, concatenated flat in return order
    float* o_read_val = out;                                     // B*R*W
    float* o_memory   = o_read_val + (size_t)BB * RR * WW;       // B*N*W
    float* o_usage    = o_memory   + (size_t)BB * NN * WW;       // B*N
    float* o_rw       = o_usage    + (size_t)BB * NN;            // B*R*N
    float* o_ww       = o_rw       + (size_t)BB * RR * NN;       // B*N
    float* o_link     = o_ww       + (size_t)BB * NN;            // B*N*N
    float* o_prec     = o_link     + (size_t)BB * NN * NN;       // B*N

    // workspace
    float* ws = (float*)d_ws;
    float* w_sum = ws;                                           // [0]
    float* w_fw  = ws + 16;                                      // B*R*N
    float* w_bw  = w_fw + (size_t)BB * RR * NN;                  // B*R*N

    // 0) zero accumulators (every call: atomics accumulate)
    k_zero<<<1, 32, 0, stream>>>(w_sum, 16);
    k_zero<<<(BB * RR * NN + 255) / 256, 256, 0, stream>>>(w_bw, BB * RR * NN);

    // 1) write head -> ww, sum_ww
    k_write_head<<<BB, 1024, 0, stream>>>(wk, wstr, ag, wg, mem, usage, o_ww, w_sum);

    // 2) memory erase/write
    k_mem_update<<<BB * NN / 8, 256, 0, stream>>>(mem, ev, wvec, o_ww, o_memory);

    // 3) temporal link update fused with fw/bw (WMMA)
    k_link<<<BB * (NN / 16), 256, 0, stream>>>(L_in, o_ww, prec, rw, o_link, w_fw, w_bw);

    // 4) read content addressing + combine + usage + precedence
    k_read_combine<<<BB, 1024, 0, stream>>>(rk, rstr, modes, fg, o_memory, usage,
                                            o_ww, prec, w_fw, w_bw, w_sum,
                                            o_rw, o_usage, o_prec);

    // 5) read values (WMMA)
    k_read_val<<<BB, 256, 0, stream>>>(o_rw, o_memory, o_read_val);
}